// DiscriminationLoss_19258633355267
// MI455X (gfx1250) — compile-verified
//
#include <hip/hip_runtime.h>
#include <hip/hip_bf16.h>
#include <math.h>

// ---- CDNA5 (gfx1250) DiscriminationLoss ----
// Pass 1: streaming segmented reduction (HBM-bound, ~5.5us floor at 23.3TB/s):
//         NT b128 loads for pred, labels staged via gfx1250 async-to-LDS
//         (double-buffered, ASYNCcnt), replicated LDS ds_add_f32 histograms.
// Pass 2: pairwise S = s_i + s_j as rank-2 matmul via v_wmma_f32_16x16x4_f32,
//         branchless validity via __ballot bitmask, operand gathers via __shfl.

typedef __attribute__((ext_vector_type(2))) float v2f;
typedef __attribute__((ext_vector_type(8))) float v8f;
typedef __attribute__((ext_vector_type(4))) float f4;
typedef __attribute__((ext_vector_type(4))) int   i4;

// exact parameter type of __builtin_amdgcn_global_load_async_to_lds_b128
// (from hipcc diagnostic: "__attribute__((__vector_size__(4 * sizeof(int)))) int *")
typedef int vi4 __attribute__((vector_size(16)));

#define NB   8
#define NC   4
#define HWSZ (896 * 896)
#define NL   33
#define NBIN 66          // [ssq: 0..32][cnt: 33..65]
#define SIGMA_DIS 3.0f

#if defined(__has_builtin)
#if __has_builtin(__builtin_amdgcn_global_load_async_to_lds_b128)
#define HAS_ASYNC 1
#endif
#endif
#ifndef HAS_ASYNC
#define HAS_ASYNC 0
#endif

#if HAS_ASYNC
#if __has_builtin(__builtin_amdgcn_s_wait_asynccnt)
#define WAIT_ASYNC_1() __builtin_amdgcn_s_wait_asynccnt(1)
#else
#define WAIT_ASYNC_1() asm volatile("s_wait_asynccnt 0x1" ::: "memory")
#endif
#endif

__global__ void dl_init(float* wsf, int* wsn, float* out) {
  int t = blockIdx.x * blockDim.x + threadIdx.x;
  if (t < NB * NBIN) wsf[t] = 0.0f;
  if (t < NB) wsn[t] = 0;
  if (t == 0) *out = 0.0f;
}

__global__ __launch_bounds__(256) void dl_hist(const float* __restrict__ pred,
                                               const int* __restrict__ labels,
                                               float* __restrict__ wsf,
                                               int* __restrict__ wsn) {
  __shared__ float hist[8][2][NBIN];   // per-wave x per-half-wave replicas
  __shared__ int smax;
#if HAS_ASYNC
  __shared__ int labbuf[2][256][4];    // double-buffered label quads
#endif
  const int tid = threadIdx.x;
  const int wv  = tid >> 5;
  const int rep = (tid >> 4) & 1;
  float* myh = &hist[wv][rep][0];
  if (tid == 0) smax = 0;
  for (int i = tid; i < 8 * 2 * NBIN; i += 256) (&hist[0][0][0])[i] = 0.0f;
  __syncthreads();

  const int b = blockIdx.y;
  const int* lab  = labels + (size_t)b * HWSZ;
  const float* pb = pred + (size_t)b * NC * HWSZ;
  const int nQ = HWSZ / 4;            // 200704 quads per image
  const int stride = gridDim.x * 256;
  const int q0 = blockIdx.x * 256 + tid;
  int maxLab = 0;

#if HAS_ASYNC
  int pbuf = 0;
  if (q0 < nQ) {
    __builtin_amdgcn_global_load_async_to_lds_b128(
        (vi4*)(lab + (q0 << 2)), (vi4*)&labbuf[0][tid][0], 0, 0);
  }
#endif

  for (int q = q0; q < nQ; q += stride) {
    const int idx = q << 2;
#if HAS_ASYNC
    // issue next label chunk before consuming current (clamped in-bounds so
    // ASYNCcnt stays uniform and no OOB access is generated)
    const int qn = q + stride;
    const int qs = (qn < nQ) ? qn : q;
    __builtin_amdgcn_global_load_async_to_lds_b128(
        (vi4*)(lab + (qs << 2)), (vi4*)&labbuf[pbuf ^ 1][tid][0], 0, 0);
#else
    i4 l4 = __builtin_nontemporal_load((const i4*)(lab + idx));
#endif
    // single-pass pred streams: non-temporal, don't churn L2
    f4 a0 = __builtin_nontemporal_load((const f4*)(pb + idx));
    f4 a1 = __builtin_nontemporal_load((const f4*)(pb + HWSZ + idx));
    f4 a2 = __builtin_nontemporal_load((const f4*)(pb + 2 * HWSZ + idx));
    f4 a3 = __builtin_nontemporal_load((const f4*)(pb + 3 * HWSZ + idx));
    const int nidx = idx + (stride << 2);   // prefetch one grid-stride ahead
    __builtin_prefetch(pb + nidx, 0, 1);
    __builtin_prefetch(pb + HWSZ + nidx, 0, 1);
    __builtin_prefetch(pb + 2 * HWSZ + nidx, 0, 1);
    __builtin_prefetch(pb + 3 * HWSZ + nidx, 0, 1);

    f4 sq = a0 * a0 + a1 * a1 + a2 * a2 + a3 * a3;  // per-pixel sum_c pred^2

#if HAS_ASYNC
    WAIT_ASYNC_1();                      // oldest async chunk has landed in LDS
    volatile int* sl = &labbuf[pbuf][tid][0];
    int lx = sl[0], ly = sl[1], lz = sl[2], lw = sl[3];
    pbuf ^= 1;
#else
    int lx = l4.x, ly = l4.y, lz = l4.z, lw = l4.w;
#endif
    atomicAdd(&myh[lx], sq.x); atomicAdd(&myh[NL + lx], 1.0f);
    atomicAdd(&myh[ly], sq.y); atomicAdd(&myh[NL + ly], 1.0f);
    atomicAdd(&myh[lz], sq.z); atomicAdd(&myh[NL + lz], 1.0f);
    atomicAdd(&myh[lw], sq.w); atomicAdd(&myh[NL + lw], 1.0f);
    maxLab = max(maxLab, max(max(lx, ly), max(lz, lw)));
  }

  atomicMax(&smax, maxLab);
  __syncthreads();
  for (int t = tid; t < NBIN; t += 256) {
    float acc = 0.0f;
    #pragma unroll
    for (int w2 = 0; w2 < 8; ++w2) acc += hist[w2][0][t] + hist[w2][1][t];
    atomicAdd(&wsf[b * NBIN + t], acc);
  }
  if (tid == 0) atomicMax(&wsn[b], smax);
}

// One wave per image. Labels 1..32 = lane+1. Validity lives in one 32-bit
// ballot mask; s-gathers are __shfl; everything branchless so EXEC is all-ones
// at each WMMA (ISA requirement). Only 3 tiles can contain i<j pairs.
__global__ __launch_bounds__(256) void dl_loss(const float* __restrict__ wsf,
                                               const int* __restrict__ wsn,
                                               float* __restrict__ out) {
  const int tid  = threadIdx.x;
  const int b    = tid >> 5;           // wave id == image id
  const int lane = tid & 31;
  const int nk   = wsn[b];
  const int lab  = lane + 1;           // this lane's label, 1..32

  float card = wsf[b * NBIN + NL + lab];
  float ssq  = wsf[b * NBIN + lab];
  float s    = (card > 0.0f) ? ssq / (card * card) : 0.0f;
  bool valid = (lab <= nk) && (card > 0.0f);
  unsigned vm = (unsigned)__ballot(valid);   // bit l => label l+1 valid

  const int half = lane >> 4;
  const int sub  = lane & 15;
  float lsum = 0.0f;

  #pragma unroll
  for (int t = 0; t < 3; ++t) {
    const int tr = (t == 2) ? 16 : 0;
    const int tc = (t == 0) ? 0 : 16;
    float srow = __shfl(s, tr + sub, 32);
    float scol = __shfl(s, tc + sub, 32);
    float lo   = (half == 0) ? 1.0f : 0.0f;
    v2f a;  a.x  = (half == 0) ? srow : 0.0f;  a.y  = lo;
    v2f bb; bb.x = lo;                         bb.y = (half == 0) ? scol : 0.0f;
    v8f c = {0.0f, 0.0f, 0.0f, 0.0f, 0.0f, 0.0f, 0.0f, 0.0f};
    v8f d = __builtin_amdgcn_wmma_f32_16x16x4_f32(
        false, a, false, bb, (short)0, c, false, false);
    #pragma unroll
    for (int r = 0; r < 8; ++r) {
      const int ri = tr + r + 8 * half;    // row label-1 (C/D: lanes16-31 -> M=8+r)
      const int ci = tc + sub;             // col label-1
      unsigned msk = (vm >> ri) & (vm >> ci) & (unsigned)(ri < ci) & 1u;
      float S  = msk ? d[r] : 1.0f;        // S_safe (keeps sqrt finite)
      float dd = SIGMA_DIS - __builtin_sqrtf(S);
      float tm = __builtin_logf(dd * dd + 1.0f);
      lsum += msk ? tm : 0.0f;
    }
  }

  #pragma unroll
  for (int off = 16; off; off >>= 1) lsum += __shfl_down(lsum, off, 32);
  if (lane == 0) {
    float nkf = (float)nk;
    float loss_b = (nk > 1) ? lsum / (nkf * (nkf - 1.0f)) : 0.0f;
    atomicAdd(out, loss_b);
  }
}

extern "C" void kernel_launch(void* const* d_in, const int* in_sizes, int n_in,
                              void* d_out, int out_size, void* d_ws, size_t ws_size,
                              hipStream_t stream) {
  (void)in_sizes; (void)n_in; (void)out_size; (void)ws_size;
  const float* pred   = (const float*)d_in[0];
  // d_in[1] (kernels_mask) is unused by the reference math.
  const int*   labels = (const int*)d_in[2];
  float* out = (float*)d_out;
  float* wsf = (float*)d_ws;                 // NB*NBIN floats
  int*   wsn = (int*)(wsf + NB * NBIN);      // NB ints (per-image max label)

  dl_init<<<1, 576, 0, stream>>>(wsf, wsn, out);
  dl_hist<<<dim3(128, NB), 256, 0, stream>>>(pred, labels, wsf, wsn);
  dl_loss<<<1, 256, 0, stream>>>(wsf, wsn, out);
}